// LOC_Attention_8194797600774
// MI455X (gfx1250) — compile-verified
//
#include <hip/hip_runtime.h>
#include <hip/hip_bf16.h>
#include <math.h>

typedef __attribute__((ext_vector_type(2))) float v2f;
typedef __attribute__((ext_vector_type(8))) float v8f;

#define GSPAT 784      // 28*28
#define JDIM  512
#define SRCDIM 1024
#define BT    128      // b*t

// ---------------------------------------------------------------------------
// Kernel 1: ctx[128,512] = context[128,1024] x W_in^T   (W_in row-major [512,1024])
// One wave32 per 16x16 output tile, K-loop of V_WMMA_F32_16X16X4_F32.
// A-frag (16x4 f32, 2 VGPR): lane&15 = M row; half=lane>>4 selects K pair.
// B-frag (4x16 f32, 2 VGPR): lane&15 = N col; half selects K pair. B[k][n]=W_in[n][k].
// ---------------------------------------------------------------------------
__global__ __launch_bounds__(256) void k_gemm_ctx(const float* __restrict__ ctxin,
                                                  const float* __restrict__ win,
                                                  float* __restrict__ ctxout) {
  const int lane = threadIdx.x & 31;
  const int wid  = threadIdx.x >> 5;
  const int tile = blockIdx.x * 8 + wid;   // 8 mtiles * 32 ntiles = 256 tiles
  const int mt = tile >> 5;
  const int nt = tile & 31;
  const int r    = lane & 15;
  const int half = lane >> 4;

  const float* arow = ctxin + (size_t)(mt * 16 + r) * SRCDIM;  // A row (M)
  const float* brow = win   + (size_t)(nt * 16 + r) * SRCDIM;  // B col n = r -> W_in row

  v8f acc = {};
  for (int k0 = 0; k0 < SRCDIM; k0 += 4) {
    v2f a, b;
    a.x = arow[k0 + 2 * half];
    a.y = arow[k0 + 2 * half + 1];
    b.x = brow[k0 + 2 * half];
    b.y = brow[k0 + 2 * half + 1];
    acc = __builtin_amdgcn_wmma_f32_16x16x4_f32(false, a, false, b, (short)0, acc,
                                                false, false);
  }
  // C/D layout: VGPR rr -> M = rr + 8*half, N = lane&15
  float* obase = ctxout + (size_t)(mt * 16) * JDIM + nt * 16;
  #pragma unroll
  for (int rr = 0; rr < 8; ++rr) {
    obase[(size_t)(rr + 8 * half) * JDIM + (lane & 15)] = acc[rr];
  }
}

// ---------------------------------------------------------------------------
// Kernel 2: G[128,784] = ctx[128,512] x fc_w   (fc_w row-major [512,784], B[k][n]=fc_w[k][n])
// ---------------------------------------------------------------------------
__global__ __launch_bounds__(256) void k_gemm_g(const float* __restrict__ ctxb,
                                                const float* __restrict__ fcw,
                                                float* __restrict__ gout) {
  const int lane = threadIdx.x & 31;
  const int wid  = threadIdx.x >> 5;
  const int tile = blockIdx.x * 8 + wid;   // 8 mtiles * 49 ntiles = 392 tiles
  const int mt = tile / 49;
  const int nt = tile % 49;
  const int r    = lane & 15;
  const int half = lane >> 4;

  const float* arow = ctxb + (size_t)(mt * 16 + r) * JDIM;
  const float* bcol = fcw + nt * 16 + r;   // column n = r, stride GSPAT over k

  v8f acc = {};
  for (int k0 = 0; k0 < JDIM; k0 += 4) {
    v2f a, b;
    a.x = arow[k0 + 2 * half];
    a.y = arow[k0 + 2 * half + 1];
    b.x = bcol[(size_t)(k0 + 2 * half) * GSPAT];
    b.y = bcol[(size_t)(k0 + 2 * half + 1) * GSPAT];
    acc = __builtin_amdgcn_wmma_f32_16x16x4_f32(false, a, false, b, (short)0, acc,
                                                false, false);
  }
  float* obase = gout + (size_t)(mt * 16) * GSPAT + nt * 16;
  #pragma unroll
  for (int rr = 0; rr < 8; ++rr) {
    obase[(size_t)(rr + 8 * half) * GSPAT + (lane & 15)] = acc[rr];
  }
}

// ---------------------------------------------------------------------------
// Block reductions (256 threads = 8 wave32)
// ---------------------------------------------------------------------------
__device__ inline float blockReduceSum256(float v, float* red) {
  const int lane = threadIdx.x & 31, wid = threadIdx.x >> 5;
  #pragma unroll
  for (int off = 16; off >= 1; off >>= 1) v += __shfl_xor(v, off, 32);
  __syncthreads();
  if (lane == 0) red[wid] = v;
  __syncthreads();
  float tot = 0.f;
  #pragma unroll
  for (int i = 0; i < 8; ++i) tot += red[i];
  return tot;
}

__device__ inline float blockReduceMax256(float v, float* red) {
  const int lane = threadIdx.x & 31, wid = threadIdx.x >> 5;
  #pragma unroll
  for (int off = 16; off >= 1; off >>= 1) v = fmaxf(v, __shfl_xor(v, off, 32));
  __syncthreads();
  if (lane == 0) red[wid] = v;
  __syncthreads();
  float m = red[0];
  #pragma unroll
  for (int i = 1; i < 8; ++i) m = fmaxf(m, red[i]);
  return m;
}

// ---------------------------------------------------------------------------
// Kernel 3: per-(b,t) fused attention. One WG of 256 threads per (b,t).
//   pass1 (HBM): s_p = ctx.cf[:,p], h_j = cf[j,:].g      (single sweep over cf)
//   softmax(s); m_p = cf[:,p].h (pass2, L2); l = s_map*m; softmax(l) -> out
// Thread t owns p in {t, t+256, t+512, t+768(if t<16)} -> s/m/softmax in registers.
// ---------------------------------------------------------------------------
__global__ __launch_bounds__(256) void k_loc_attn(const float* __restrict__ curr,
                                                  const float* __restrict__ ctxb,
                                                  const float* __restrict__ gbuf,
                                                  float* __restrict__ out) {
  __shared__ float ls_ctx[JDIM];
  __shared__ float ls_g[GSPAT];
  __shared__ float ls_h[JDIM];
  __shared__ float ls_hpart[JDIM * 8];
  __shared__ float ls_red[8];

  const int it   = blockIdx.x;
  const int t    = threadIdx.x;
  const int lane = t & 31, wid = t >> 5;
  const float* cf = curr + (size_t)it * JDIM * GSPAT;

  ls_ctx[t]       = ctxb[(size_t)it * JDIM + t];
  ls_ctx[t + 256] = ctxb[(size_t)it * JDIM + t + 256];
  ls_g[t]         = gbuf[(size_t)it * GSPAT + t];
  ls_g[t + 256]   = gbuf[(size_t)it * GSPAT + t + 256];
  ls_g[t + 512]   = gbuf[(size_t)it * GSPAT + t + 512];
  if (t < 16) ls_g[t + 768] = gbuf[(size_t)it * GSPAT + t + 768];
  __syncthreads();

  const bool has4 = (t < 16);
  const float g0 = ls_g[t], g1 = ls_g[t + 256], g2 = ls_g[t + 512];
  const float g3 = has4 ? ls_g[t + 768] : 0.f;

  // ---- pass 1: s (register) and h partials, one read of cf from HBM ----
  float s0 = 0.f, s1 = 0.f, s2 = 0.f, s3 = 0.f;
  for (int j = 0; j < JDIM; ++j) {
    const float* row = cf + (size_t)j * GSPAT;
    if (j < JDIM - 1) __builtin_prefetch(row + GSPAT + t, 0, 3);  // next row
    const float c0 = row[t];
    const float c1 = row[t + 256];
    const float c2 = row[t + 512];
    const float c3 = has4 ? row[t + 768] : 0.f;
    const float cj = ls_ctx[j];
    s0 = fmaf(cj, c0, s0); s1 = fmaf(cj, c1, s1);
    s2 = fmaf(cj, c2, s2); s3 = fmaf(cj, c3, s3);
    float hp = c0 * g0 + c1 * g1 + c2 * g2 + c3 * g3;
    #pragma unroll
    for (int off = 16; off >= 1; off >>= 1) hp += __shfl_xor(hp, off, 32);
    if (lane == 0) ls_hpart[j * 8 + wid] = hp;
  }
  __syncthreads();
  for (int j = t; j < JDIM; j += 256) {
    float h = 0.f;
    #pragma unroll
    for (int w = 0; w < 8; ++w) h += ls_hpart[j * 8 + w];
    ls_h[j] = h;
  }
  // (barriers inside the reductions below publish ls_h before pass 2)

  // ---- softmax over s (784), kept in registers ----
  float mx = fmaxf(fmaxf(s0, s1), s2);
  if (has4) mx = fmaxf(mx, s3);
  const float M = blockReduceMax256(mx, ls_red);
  const float e0 = __expf(s0 - M), e1 = __expf(s1 - M), e2 = __expf(s2 - M);
  const float e3 = has4 ? __expf(s3 - M) : 0.f;
  const float denom = blockReduceSum256(e0 + e1 + e2 + e3, ls_red);
  const float inv = 1.0f / denom;
  const float sm0 = e0 * inv, sm1 = e1 * inv, sm2 = e2 * inv, sm3 = e3 * inv;

  // ---- pass 2: m_p = sum_j cf[j,p] * h_j (cf now L2-resident) ----
  float m0 = 0.f, m1 = 0.f, m2 = 0.f, m3 = 0.f;
  for (int j = 0; j < JDIM; ++j) {
    const float* row = cf + (size_t)j * GSPAT;
    const float hj = ls_h[j];
    m0 = fmaf(hj, row[t], m0);
    m1 = fmaf(hj, row[t + 256], m1);
    m2 = fmaf(hj, row[t + 512], m2);
    if (has4) m3 = fmaf(hj, row[t + 768], m3);
  }

  // ---- final logits l = s_map * m ; softmax(l) -> output ----
  const float l0 = sm0 * m0, l1 = sm1 * m1, l2 = sm2 * m2;
  const float l3 = has4 ? sm3 * m3 : -INFINITY;
  float lmx = fmaxf(fmaxf(l0, l1), fmaxf(l2, l3));
  const float LM = blockReduceMax256(lmx, ls_red);
  const float f0 = __expf(l0 - LM), f1 = __expf(l1 - LM), f2 = __expf(l2 - LM);
  const float f3 = has4 ? __expf(l3 - LM) : 0.f;
  const float S = blockReduceSum256(f0 + f1 + f2 + f3, ls_red);
  const float invS = 1.0f / S;

  float* o = out + (size_t)it * GSPAT;
  o[t] = f0 * invS;
  o[t + 256] = f1 * invS;
  o[t + 512] = f2 * invS;
  if (has4) o[t + 768] = f3 * invS;
}

// ---------------------------------------------------------------------------
extern "C" void kernel_launch(void* const* d_in, const int* in_sizes, int n_in,
                              void* d_out, int out_size, void* d_ws, size_t ws_size,
                              hipStream_t stream) {
  const float* context = (const float*)d_in[0];   // [8,16,1024]
  const float* curr    = (const float*)d_in[1];   // [8,16,512,784]
  const float* win     = (const float*)d_in[2];   // [512,1024]
  const float* fcw     = (const float*)d_in[3];   // [512,784]
  // d_in[4] = fc_b: constant additive shift across p -> cancels in softmax (and is zero).

  float* ctxbuf = (float*)d_ws;                   // [128,512]  (256 KB)
  float* gbuf   = ctxbuf + (size_t)BT * JDIM;     // [128,784]  (~401 KB)

  k_gemm_ctx<<<32, 256, 0, stream>>>(context, win, ctxbuf);   // 256 tiles / 8 waves
  k_gemm_g  <<<49, 256, 0, stream>>>(ctxbuf, fcw, gbuf);      // 392 tiles / 8 waves
  k_loc_attn<<<BT, 256, 0, stream>>>(curr, ctxbuf, gbuf, (float*)d_out);
}